// SimGCL_Encoder_9990093931258
// MI455X (gfx1250) — compile-verified
//
#include <hip/hip_runtime.h>
#include <math.h>

// ---- Problem constants (match reference) ----
#define USER_NUM 100000
#define ITEM_NUM 50000
#define NN       (USER_NUM + ITEM_NUM)   // 150000 (multiple of 16)
#define EMB      64
#define NNZT     4800000
#define NLAYERS  3
#define EPSV     0.1f

typedef __attribute__((ext_vector_type(2))) float v2f;
typedef __attribute__((ext_vector_type(8))) float v8f;

// ---------------------------------------------------------------------------
// Kernel 1: ego = concat(user_emb, item_emb); out = 0.25 * ego
// ---------------------------------------------------------------------------
__global__ void simgcl_init(const float* __restrict__ user,
                            const float* __restrict__ item,
                            float* __restrict__ ego,
                            float* __restrict__ out) {
  size_t i = (size_t)blockIdx.x * blockDim.x + threadIdx.x;   // float4 index
  const size_t total = (size_t)NN * EMB / 4;
  if (i >= total) return;
  const size_t userQ = (size_t)USER_NUM * EMB / 4;
  float4 v = (i < userQ) ? ((const float4*)user)[i]
                         : ((const float4*)item)[i - userQ];
  ((float4*)ego)[i] = v;
  float4 o = {0.25f * v.x, 0.25f * v.y, 0.25f * v.z, 0.25f * v.w};
  ((float4*)out)[i] = o;
}

// ---------------------------------------------------------------------------
// Kernel 2: zero the scatter target
// ---------------------------------------------------------------------------
__global__ void simgcl_zero(float* __restrict__ y) {
  size_t i = (size_t)blockIdx.x * blockDim.x + threadIdx.x;
  const size_t total = (size_t)NN * EMB / 4;
  if (i >= total) return;
  float4 z = {0.f, 0.f, 0.f, 0.f};
  ((float4*)y)[i] = z;
}

// ---------------------------------------------------------------------------
// Kernel 3: SpMM via gather + native f32 atomic scatter (L2-resident RMW).
// Each wave loads 32 edges coalesced, broadcasts (row,col,val) per edge,
// then all 32 lanes move one 256B embedding row (float2/lane).
// ---------------------------------------------------------------------------
__global__ void simgcl_spmm(const int*   __restrict__ rows,
                            const int*   __restrict__ cols,
                            const float* __restrict__ vals,
                            const float* __restrict__ x,
                            float*       __restrict__ y) {
  const int lane = threadIdx.x & 31;
  const int wid  = blockIdx.x * (blockDim.x >> 5) + (threadIdx.x >> 5);
  const int nw   = gridDim.x * (blockDim.x >> 5);
  const int stride = nw * 32;

  for (int base = wid * 32; base < NNZT; base += stride) {
    int e = base + lane;
    int r = 0, c = 0;
    float v = 0.f;
    if (e < NNZT) {
      r = rows[e];
      c = cols[e];
      v = vals[e];
    }
    // Prefetch next sweep's edge data (emits global_prefetch_b8).
    int pn = base + stride + lane;
    if (pn < NNZT) {
      __builtin_prefetch((const void*)(rows + pn), 0, 1);
      __builtin_prefetch((const void*)(cols + pn), 0, 1);
      __builtin_prefetch((const void*)(vals + pn), 0, 1);
    }
    int cnt = NNZT - base;
    if (cnt > 32) cnt = 32;
    for (int j = 0; j < cnt; ++j) {
      const int   rj = __shfl(r, j, 32);
      const int   cj = __shfl(c, j, 32);
      const float vj = __shfl(v, j, 32);
      const float2 xv = *(const float2*)(x + (size_t)cj * EMB + lane * 2);
      float* dst = y + (size_t)rj * EMB + lane * 2;
      // Native FP32 atomics (global_atomic_add_f32), resolved in the 192MB L2.
      __hip_atomic_fetch_add(dst + 0, vj * xv.x, __ATOMIC_RELAXED,
                             __HIP_MEMORY_SCOPE_AGENT);
      __hip_atomic_fetch_add(dst + 1, vj * xv.y, __ATOMIC_RELAXED,
                             __HIP_MEMORY_SCOPE_AGENT);
    }
  }
}

// ---------------------------------------------------------------------------
// Kernel 4: noise perturbation + accumulate mean.
//   ego = ego + sign(ego) * (noise_row / (||noise_row|| + 1e-12)) * EPS
//   out += 0.25 * ego
// Row norms (16 rows / wave-tile) computed exactly in f32 with
// V_WMMA_F32_16X16X4_F32:  D = T * T^T accumulated over K=64 in 16 chunks;
// diag(D) = squared norms. With the 16x4 f32 A layout, the matching
// B = T^T fragment is register-identical to A, so A feeds both operands.
// ---------------------------------------------------------------------------
__global__ void simgcl_noise_acc(float*       __restrict__ ego,  // spmm result -> perturbed
                                 const float* __restrict__ nz,   // this layer's noise [N,64]
                                 float*       __restrict__ out) {
  __shared__ float norms[8 * 16];                 // 8 waves/block * 16 rows
  const int lane = threadIdx.x & 31;
  const int slot = threadIdx.x >> 5;
  const int wt   = blockIdx.x * 8 + slot;         // 16-row tile index
  const int NT   = NN / 16;                       // 9375 tiles exactly
  const bool active = (wt < NT);                  // wave-uniform -> EXEC all-1s inside
  const int r0 = wt * 16;

  if (active) {
    const int m  = lane & 15;                     // row within tile
    const int hb = lane >> 4;                     // K half-select
    const float* tp = nz + (size_t)(r0 + m) * EMB + 2 * hb;
    v8f cacc = {0.f, 0.f, 0.f, 0.f, 0.f, 0.f, 0.f, 0.f};
#pragma unroll
    for (int q = 0; q < 16; ++q) {                // K = 64 in chunks of 4
      const float2 t = *(const float2*)(tp + 4 * q);
      v2f a;
      a.x = t.x;
      a.y = t.y;
      // D += A(16x4) * A^T(4x16), all f32 (exact Gram accumulate)
      cacc = __builtin_amdgcn_wmma_f32_16x16x4_f32(
          false, a, false, a, (short)0, cacc, false, false);
    }
    // Diagonal of C: lanes 0-7 hold diag m=lane in c[lane];
    // lanes 24-31 hold diag m=lane-16 in c[lane-24].
    const int n = (lane < 16) ? lane : (lane - 16);
#pragma unroll
    for (int v = 0; v < 8; ++v) {
      const int mm = (lane < 16) ? v : (v + 8);
      if (mm == n) norms[slot * 16 + mm] = cacc[v];
    }
  }
  __syncthreads();
  if (active) {
#pragma unroll 4
    for (int rr = 0; rr < 16; ++rr) {
      const float nsq   = norms[slot * 16 + rr];
      const float scale = EPSV / (sqrtf(nsq) + 1e-12f);
      const size_t off  = (size_t)(r0 + rr) * EMB + lane * 2;
      const float2 t  = *(const float2*)(ego + off);
      const float2 nk = *(const float2*)(nz + off);
      const float sx = (t.x > 0.f) ? 1.f : ((t.x < 0.f) ? -1.f : 0.f);
      const float sy = (t.y > 0.f) ? 1.f : ((t.y < 0.f) ? -1.f : 0.f);
      float2 e;
      e.x = t.x + sx * nk.x * scale;
      e.y = t.y + sy * nk.y * scale;
      *(float2*)(ego + off) = e;
      float2 o = *(float2*)(out + off);
      o.x += 0.25f * e.x;
      o.y += 0.25f * e.y;
      *(float2*)(out + off) = o;
    }
  }
}

// ---------------------------------------------------------------------------
// Host orchestration (graph-capture safe: only kernel launches on `stream`)
// ---------------------------------------------------------------------------
extern "C" void kernel_launch(void* const* d_in, const int* in_sizes, int n_in,
                              void* d_out, int out_size, void* d_ws, size_t ws_size,
                              hipStream_t stream) {
  const float* user  = (const float*)d_in[0];
  const float* item  = (const float*)d_in[1];
  const int*   arow  = (const int*)d_in[2];
  const int*   acol  = (const int*)d_in[3];
  const float* aval  = (const float*)d_in[4];
  const float* noise = (const float*)d_in[5];
  float* out = (float*)d_out;

  // Ping-pong ego buffers in workspace: 2 * N*EMB*4B = 76.8 MB
  float* bufA = (float*)d_ws;
  float* bufB = bufA + (size_t)NN * EMB;

  const int tb = 256;
  const size_t quads = (size_t)NN * EMB / 4;
  const int vecBlocks = (int)((quads + tb - 1) / tb);
  const int spmmBlocks  = 4096;                  // grid-stride over 4.8M edges
  const int noiseBlocks = (NN / 16 + 7) / 8;     // 8 wave-tiles per block

  simgcl_init<<<vecBlocks, tb, 0, stream>>>(user, item, bufA, out);

  for (int k = 0; k < NLAYERS; ++k) {
    simgcl_zero<<<vecBlocks, tb, 0, stream>>>(bufB);
    simgcl_spmm<<<spmmBlocks, tb, 0, stream>>>(arow, acol, aval, bufA, bufB);
    simgcl_noise_acc<<<noiseBlocks, tb, 0, stream>>>(
        bufB, noise + (size_t)k * NN * EMB, out);
    float* t = bufA; bufA = bufB; bufB = t;
  }
}